// HeteroLayerCausalUni_51058571214890
// MI455X (gfx1250) — compile-verified
//
#include <hip/hip_runtime.h>

typedef __attribute__((ext_vector_type(2))) float v2f;
typedef __attribute__((ext_vector_type(8))) float v8f;

#define NW 30000
#define NT 2000
#define ND 20000
#define DD 128

#define E_WW 600000
#define E_WT 300000
#define E_WD 400000
#define E_TD 100000
#define E_TT 40000

// ---------------------------------------------------------------------------
// Y[nrows,128] = X[nrows,128] @ W[128,128] (+ bias)
// One 256-thread block (8 waves) computes a 16x128 output stripe.
// Wave w handles columns [16w, 16w+16) with V_WMMA_F32_16X16X4_F32, K-loop of 32.
// Fragment layouts per CDNA5 ISA 7.12.2 (fp32 16x16x4):
//   A (v2f): lane L -> row m0+(L&15), K = 2*(L>>4) + {0,1}
//   B (v2f): lane L -> col n0+(L&15), K = 2*(L>>4) + {0,1}
//   C/D (v8f): elem v -> row m0 + v + 8*(L>>4), col n0+(L&15)
// nrows must be a multiple of 16 (30000/2000/20000 all are).
// ---------------------------------------------------------------------------
__global__ __launch_bounds__(256) void k_gemm128(const float* __restrict__ X,
                                                 const float* __restrict__ W,
                                                 const float* __restrict__ bias,
                                                 float* __restrict__ Y, int nrows) {
  const int lane = threadIdx.x & 31;
  const int wave = threadIdx.x >> 5;
  const int m0 = blockIdx.x << 4;
  const int n0 = wave << 4;
  const int mr = lane & 15;
  const int kh = lane >> 4;  // 0 or 1
  const float* xrow = X + (size_t)(m0 + mr) * DD;
  const int col = n0 + mr;

  v8f acc = {};
#pragma unroll
  for (int k = 0; k < DD; k += 4) {
    const int ka = k + 2 * kh;
    v2f a, b;
    a[0] = xrow[ka];
    a[1] = xrow[ka + 1];
    b[0] = W[(size_t)ka * DD + col];
    b[1] = W[(size_t)(ka + 1) * DD + col];
    acc = __builtin_amdgcn_wmma_f32_16x16x4_f32(false, a, false, b,
                                                (short)0, acc, false, false);
  }
  const float bv = bias ? bias[col] : 0.0f;
#pragma unroll
  for (int v = 0; v < 8; ++v) {
    const int row = m0 + v + 8 * kh;
    Y[(size_t)row * DD + col] = acc[v] + bv;
  }
}

// ---------------------------------------------------------------------------
// Edge scatter: one wave per edge; lane owns 4 contiguous floats of the
// 128-float feature. acc[dst] += Wh[src]*w via global_atomic_add_f32;
// lane 0 counts degree. Accumulators stay resident in the 192MB L2.
// ---------------------------------------------------------------------------
__global__ __launch_bounds__(256) void k_scatter(const float* __restrict__ Wh,
                                                 const int* __restrict__ src,
                                                 const int* __restrict__ dst,
                                                 const float* __restrict__ w,
                                                 float* __restrict__ acc,
                                                 float* __restrict__ deg,
                                                 int nedges) {
  const int e = blockIdx.x * 8 + (threadIdx.x >> 5);
  if (e >= nedges) return;
  const int lane = threadIdx.x & 31;
  const int s = src[e];
  const int d = dst[e];
  const float ww = w[e];
  const float4 v = ((const float4*)(Wh + (size_t)s * DD))[lane];
  float* out = acc + (size_t)d * DD + lane * 4;
  atomicAdd(out + 0, v.x * ww);
  atomicAdd(out + 1, v.y * ww);
  atomicAdd(out + 2, v.z * ww);
  atomicAdd(out + 3, v.w * ww);
  if (lane == 0) atomicAdd(deg + d, 1.0f);
}

__global__ void k_zero(float* __restrict__ p, int n) {
  const int i = blockIdx.x * blockDim.x + threadIdx.x;
  if (i < n) p[i] = 0.0f;
}

// h = deg>0 ? acc/deg : 0
__global__ void k_norm1(const float* __restrict__ acc, const float* __restrict__ deg,
                        float* __restrict__ h, int n) {
  const int i = blockIdx.x * blockDim.x + threadIdx.x;
  if (i >= n) return;
  const float dg = deg[i >> 7];
  h[i] = dg > 0.0f ? acc[i] / dg : 0.0f;
}

// h = mean_rel1 + mean_rel2 (cross-relation sum reducer)
__global__ void k_norm2(const float* __restrict__ a1, const float* __restrict__ d1,
                        const float* __restrict__ a2, const float* __restrict__ d2,
                        float* __restrict__ h, int n) {
  const int i = blockIdx.x * blockDim.x + threadIdx.x;
  if (i >= n) return;
  const int r = i >> 7;
  const float g1 = d1[r];
  const float g2 = d2[r];
  float v = 0.0f;
  if (g1 > 0.0f) v += a1[i] / g1;
  if (g2 > 0.0f) v += a2[i] / g2;
  h[i] = v;
}

// out = P + causal*Pc - (bern * !causal)*Pn   (row masks commute out of GEMMs)
__global__ void k_topic_combine(const float* __restrict__ P, const float* __restrict__ Pc,
                                const float* __restrict__ Pn,
                                const float* __restrict__ effect,
                                const float* __restrict__ bern,
                                float* __restrict__ out, int n) {
  const int i = blockIdx.x * blockDim.x + threadIdx.x;
  if (i >= n) return;
  const int r = i >> 7;
  const float e = effect[r];
  const float causal = (e != 0.0f) ? 1.0f : 0.0f;
  const float rm = bern[r] * (1.0f - causal);
  out[i] = P[i] + causal * Pc[i] - rm * Pn[i];
}

static inline void launch_gemm(const float* X, const float* W, const float* b,
                               float* Y, int nrows, hipStream_t stream) {
  k_gemm128<<<nrows / 16, 256, 0, stream>>>(X, W, b, Y, nrows);
}

extern "C" void kernel_launch(void* const* d_in, const int* in_sizes, int n_in,
                              void* d_out, int out_size, void* d_ws, size_t ws_size,
                              hipStream_t stream) {
  (void)in_sizes; (void)n_in; (void)out_size; (void)ws_size;

  const float* feat_word  = (const float*)d_in[0];
  const float* feat_topic = (const float*)d_in[1];
  const float* effect     = (const float*)d_in[2];
  const float* bern_td    = (const float*)d_in[3];
  const float* bern_tt    = (const float*)d_in[4];
  const int*   src_ww = (const int*)d_in[5];
  const int*   dst_ww = (const int*)d_in[6];
  const float* w_ww   = (const float*)d_in[7];
  const int*   src_wt = (const int*)d_in[8];
  const int*   dst_wt = (const int*)d_in[9];
  const float* w_wt   = (const float*)d_in[10];
  const int*   src_wd = (const int*)d_in[11];
  const int*   dst_wd = (const int*)d_in[12];
  const float* w_wd   = (const float*)d_in[13];
  const int*   src_td = (const int*)d_in[14];
  const int*   dst_td = (const int*)d_in[15];
  const float* w_td   = (const float*)d_in[16];
  const int*   src_tt = (const int*)d_in[17];
  const int*   dst_tt = (const int*)d_in[18];
  const float* w_tt   = (const float*)d_in[19];
  const float* W_ww = (const float*)d_in[20]; const float* b_ww = (const float*)d_in[21];
  const float* W_wt = (const float*)d_in[22]; const float* b_wt = (const float*)d_in[23];
  const float* W_wd = (const float*)d_in[24]; const float* b_wd = (const float*)d_in[25];
  const float* W_td = (const float*)d_in[26]; const float* b_td = (const float*)d_in[27];
  const float* W_tt = (const float*)d_in[28]; const float* b_tt = (const float*)d_in[29];
  const float* W_causal = (const float*)d_in[30];
  const float* W_noise  = (const float*)d_in[31];

  // Workspace layout (floats). acc_w..deg_w and acc_wt..deg_td are contiguous
  // so each can be zeroed with a single fill.
  float* ws     = (float*)d_ws;
  float* Wh_ww  = ws;
  float* acc_w  = Wh_ww  + (size_t)NW * DD;
  float* deg_w  = acc_w  + (size_t)NW * DD;
  float* Wh_wt  = deg_w  + NW;
  float* Wh_wd  = Wh_wt  + (size_t)NW * DD;
  float* P_td   = Wh_wd  + (size_t)NW * DD;
  float* P_tt   = P_td   + (size_t)NT * DD;
  float* P_c    = P_tt   + (size_t)NT * DD;
  float* P_n    = P_c    + (size_t)NT * DD;
  float* Wh_td  = P_n    + (size_t)NT * DD;
  float* Wh_tt  = Wh_td  + (size_t)NT * DD;
  float* acc_wt = Wh_tt  + (size_t)NT * DD;
  float* acc_tt = acc_wt + (size_t)NT * DD;
  float* deg_wt = acc_tt + (size_t)NT * DD;
  float* deg_tt = deg_wt + NT;
  float* acc_wd = deg_tt + NT;
  float* acc_td = acc_wd + (size_t)ND * DD;
  float* deg_wd = acc_td + (size_t)ND * DD;
  float* deg_td = deg_wd + ND;

  float* h_word  = (float*)d_out;
  float* h_topic = h_word  + (size_t)NW * DD;
  float* h_doc   = h_topic + (size_t)NT * DD;

  const int TB = 256;

  // ---- Phase 1: word->word -------------------------------------------------
  {
    const int nz = NW * DD + NW;  // acc_w + deg_w contiguous
    k_zero<<<(nz + TB - 1) / TB, TB, 0, stream>>>(acc_w, nz);
  }
  launch_gemm(feat_word, W_ww, b_ww, Wh_ww, NW, stream);
  k_scatter<<<(E_WW + 7) / 8, TB, 0, stream>>>(Wh_ww, src_ww, dst_ww, w_ww,
                                               acc_w, deg_w, E_WW);
  k_norm1<<<(NW * DD + TB - 1) / TB, TB, 0, stream>>>(acc_w, deg_w, h_word, NW * DD);

  // ---- Phase 2: projections ------------------------------------------------
  launch_gemm(h_word, W_wt, b_wt, Wh_wt, NW, stream);
  launch_gemm(h_word, W_wd, b_wd, Wh_wd, NW, stream);
  launch_gemm(feat_topic, W_td, b_td, P_td, NT, stream);
  launch_gemm(feat_topic, W_tt, b_tt, P_tt, NT, stream);
  launch_gemm(feat_topic, W_causal, nullptr, P_c, NT, stream);
  launch_gemm(feat_topic, W_noise,  nullptr, P_n, NT, stream);
  k_topic_combine<<<(NT * DD + TB - 1) / TB, TB, 0, stream>>>(
      P_td, P_c, P_n, effect, bern_td, Wh_td, NT * DD);
  k_topic_combine<<<(NT * DD + TB - 1) / TB, TB, 0, stream>>>(
      P_tt, P_c, P_n, effect, bern_tt, Wh_tt, NT * DD);

  // ---- Phase 2: aggregations ----------------------------------------------
  {
    const int nz = 2 * NT * DD + 2 * NT + 2 * ND * DD + 2 * ND;  // acc_wt..deg_td
    k_zero<<<(nz + TB - 1) / TB, TB, 0, stream>>>(acc_wt, nz);
  }
  k_scatter<<<(E_WT + 7) / 8, TB, 0, stream>>>(Wh_wt, src_wt, dst_wt, w_wt,
                                               acc_wt, deg_wt, E_WT);
  k_scatter<<<(E_TT + 7) / 8, TB, 0, stream>>>(Wh_tt, src_tt, dst_tt, w_tt,
                                               acc_tt, deg_tt, E_TT);
  k_scatter<<<(E_WD + 7) / 8, TB, 0, stream>>>(Wh_wd, src_wd, dst_wd, w_wd,
                                               acc_wd, deg_wd, E_WD);
  k_scatter<<<(E_TD + 7) / 8, TB, 0, stream>>>(Wh_td, src_td, dst_td, w_td,
                                               acc_td, deg_td, E_TD);

  k_norm2<<<(NT * DD + TB - 1) / TB, TB, 0, stream>>>(
      acc_wt, deg_wt, acc_tt, deg_tt, h_topic, NT * DD);
  k_norm2<<<(ND * DD + TB - 1) / TB, TB, 0, stream>>>(
      acc_wd, deg_wd, acc_td, deg_td, h_doc, ND * DD);
}